// EchoStateNetwork_154618822842
// MI455X (gfx1250) — compile-verified
//
#include <hip/hip_runtime.h>

#define N_NODES  4000
#define N_IN     32
#define N_OUT    32
#define T_STEPS  2048
#define KEXT     64     // N_IN + N_OUT
#define KOUT     4032   // N_IN + N_NODES
#define LEAK     0.99f

typedef __attribute__((ext_vector_type(2))) float v2f;
typedef __attribute__((ext_vector_type(8))) float v8f;

// ---------------------------------------------------------------------------
// Phase 1: drive[T][N] = [X|y] @ [W_in|W_fb]^T + b     (f32 WMMA 16x16x4)
// ---------------------------------------------------------------------------
__global__ void drive_gemm_kernel(const float* __restrict__ X,
                                  const float* __restrict__ Y,
                                  const float* __restrict__ Win,
                                  const float* __restrict__ Wfb,
                                  const float* __restrict__ bptr,
                                  float* __restrict__ drive) {
    const int lane = threadIdx.x & 31;
    const int wave = (blockIdx.x * blockDim.x + threadIdx.x) >> 5;
    const int tilesN = N_NODES / 16;              // 250
    const int tM = wave / tilesN;                 // 0..127
    const int tN = wave % tilesN;                 // 0..249
    const int t0 = tM * 16, n0 = tN * 16;
    const int half = lane >> 4;
    const int l = lane & 15;
    const int arow = t0 + l;
    const int bcol = n0 + l;
    const float bias = bptr[0];

    v8f c = {};
#pragma unroll
    for (int kb = 0; kb < KEXT / 4; ++kb) {
        const int k = kb * 4 + 2 * half;
        v2f a, bf;
        if (kb < 8) {                             // K in [0,32): X / W_in
            a  = *(const v2f*)(X   + arow * N_IN + k);
            bf = *(const v2f*)(Win + bcol * N_IN + k);
        } else {                                  // K in [32,64): y / W_fb
            a  = *(const v2f*)(Y   + arow * N_OUT + (k - N_IN));
            bf = *(const v2f*)(Wfb + bcol * N_OUT + (k - N_IN));
        }
        c = __builtin_amdgcn_wmma_f32_16x16x4_f32(false, a, false, bf,
                                                  (short)0, c, false, false);
    }
#pragma unroll
    for (int r = 0; r < 8; ++r)
        drive[(size_t)(t0 + r + 8 * half) * N_NODES + n0 + l] = c[r] + bias;
}

// ---------------------------------------------------------------------------
// One-time W_res f32 -> bf16 pack (RNE). Halves the dominant per-step L2
// traffic of the sequential phase (131 GB -> 65 GB over 2048 steps).
// ---------------------------------------------------------------------------
__device__ __forceinline__ unsigned f2bf_rne(float f) {
    unsigned b = __float_as_uint(f);
    return (b + 0x7fffu + ((b >> 16) & 1u)) >> 16;
}

__global__ void wres_to_bf16_kernel(const float* __restrict__ W,
                                    unsigned* __restrict__ Wb) {
    const size_t i = (size_t)blockIdx.x * blockDim.x + threadIdx.x; // 4M threads
    const float4 f = *(const float4*)(W + 4 * i);
    uint2 o;
    o.x = f2bf_rne(f.x) | (f2bf_rne(f.y) << 16);
    o.y = f2bf_rne(f.z) | (f2bf_rne(f.w) << 16);
    *(uint2*)(Wb + 2 * i) = o;
}

// ---------------------------------------------------------------------------
// Phase 2: persistent sequential reservoir.
// h_{t+1} = LEAK*tanh(drive_t + W_res h_t) + (1-LEAK)*h_t ; states[t] = h_{t+1}
// 125 blocks x 256 threads; block owns 32 rows; 8 lanes per row.
// h staged into LDS with gfx1250 async global->LDS copies (ASYNCcnt).
// Weights streamed from L2 as bf16 (fast path) or f32 (fallback).
// ---------------------------------------------------------------------------
#define RES_BLOCKS  125
#define RES_THREADS 256

template <bool USE_BF16>
__launch_bounds__(RES_THREADS, 1)
__global__ void reservoir_kernel(const float*    __restrict__ Wres_f32,
                                 const unsigned* __restrict__ Wres_bf16,
                                 const float* __restrict__ drive,
                                 float* __restrict__ h0,
                                 float* __restrict__ h1,
                                 float* __restrict__ states,
                                 unsigned* __restrict__ barCount,
                                 unsigned* __restrict__ barGen) {
    __shared__ __align__(16) float lds_h[N_NODES];
    const int tid = threadIdx.x;
    const int row = blockIdx.x * 32 + (tid >> 3);  // 32 rows / block
    const int g   = tid & 7;                       // 8 lanes per row

    const float*    __restrict__ wrow_f  = Wres_f32  + (size_t)row * N_NODES;
    const unsigned* __restrict__ wrow_bf = Wres_bf16 + (size_t)row * (N_NODES / 2);

    // low 32 bits of a generic LDS pointer == LDS byte offset (aperture rule)
    const unsigned lds_base = (unsigned)(uintptr_t)(&lds_h[0]);

    float* hb[2] = { h0, h1 };
    for (int t = 0; t < T_STEPS; ++t) {
        const float* hin  = hb[t & 1];
        float*       hout = hb[(t + 1) & 1];

        // ---- stage h (16 KB) into LDS via async global->LDS b128 copies ----
        for (int c = tid; c < N_NODES / 4; c += RES_THREADS) {  // 1000 chunks
            const unsigned off = (unsigned)c * 16u;
            asm volatile("global_load_async_to_lds_b128 %0, %1, %2 offset:0"
                         :: "v"(lds_base + off), "v"(off), "s"(hin)
                         : "memory");
        }
        asm volatile("s_wait_asynccnt 0x0" ::: "memory");
        __syncthreads();

        // ---- partial dot: lane g covers k = 4*g + 32*j ----
        float acc = 0.f;
        if (USE_BF16) {
#pragma unroll 5
            for (int k = g * 4; k < N_NODES; k += 32) {
                const uint2 w = *(const uint2*)(wrow_bf + (k >> 1)); // 4 bf16
                const float4 h4 = *(const float4*)(&lds_h[k]);
                acc += __uint_as_float(w.x << 16)          * h4.x;
                acc += __uint_as_float(w.x & 0xffff0000u)  * h4.y;
                acc += __uint_as_float(w.y << 16)          * h4.z;
                acc += __uint_as_float(w.y & 0xffff0000u)  * h4.w;
            }
        } else {
#pragma unroll 5
            for (int k = g * 4; k < N_NODES; k += 32) {
                const float4 w  = *(const float4*)(wrow_f + k);
                const float4 h4 = *(const float4*)(&lds_h[k]);
                acc += w.x * h4.x + w.y * h4.y + w.z * h4.z + w.w * h4.w;
            }
        }
        // reduce 8 lanes -> lane g==0 of each row group
        acc += __shfl_xor(acc, 1);
        acc += __shfl_xor(acc, 2);
        acc += __shfl_xor(acc, 4);

        if (g == 0) {
            const float pre = acc + drive[(size_t)t * N_NODES + row];
            const float hn  = LEAK * tanhf(pre) + (1.0f - LEAK) * lds_h[row];
            hout[row] = hn;
            states[(size_t)t * N_NODES + row] = hn;
        }
        __syncthreads();   // LDS reads done before next-iteration restage

        // ---- device-wide barrier (sense = step index) ----
        if (tid == 0) {
            const unsigned target = (unsigned)(t + 1);
            unsigned prev = __hip_atomic_fetch_add(barCount, 1u,
                                __ATOMIC_ACQ_REL, __HIP_MEMORY_SCOPE_AGENT);
            if (prev == (unsigned)(gridDim.x - 1)) {
                __hip_atomic_store(barCount, 0u,
                    __ATOMIC_RELAXED, __HIP_MEMORY_SCOPE_AGENT);
                __hip_atomic_store(barGen, target,
                    __ATOMIC_RELEASE, __HIP_MEMORY_SCOPE_AGENT);
            } else {
                while (__hip_atomic_load(barGen,
                        __ATOMIC_ACQUIRE, __HIP_MEMORY_SCOPE_AGENT) < target) {
                    __builtin_amdgcn_s_sleep(2);
                }
            }
        }
        __syncthreads();
    }
}

// ---------------------------------------------------------------------------
// Phase 3: out[T][32] = [X | states] @ W_out^T          (f32 WMMA 16x16x4)
// ---------------------------------------------------------------------------
__global__ void out_gemm_kernel(const float* __restrict__ X,
                                const float* __restrict__ states,
                                const float* __restrict__ Wout,
                                float* __restrict__ out) {
    const int lane = threadIdx.x & 31;
    const int wave = (blockIdx.x * blockDim.x + threadIdx.x) >> 5;
    const int tilesN = N_OUT / 16;                // 2
    const int tM = wave / tilesN;                 // 0..127
    const int tN = wave % tilesN;
    const int t0 = tM * 16, n0 = tN * 16;
    const int half = lane >> 4;
    const int l = lane & 15;
    const int arow = t0 + l;
    const int bcol = n0 + l;

    v8f c = {};
    for (int kb = 0; kb < KOUT / 4; ++kb) {
        const int k = kb * 4 + 2 * half;
        v2f a;
        if (kb < 8)  a = *(const v2f*)(X + arow * N_IN + k);
        else         a = *(const v2f*)(states + (size_t)arow * N_NODES + (k - N_IN));
        v2f bf = *(const v2f*)(Wout + bcol * KOUT + k);
        c = __builtin_amdgcn_wmma_f32_16x16x4_f32(false, a, false, bf,
                                                  (short)0, c, false, false);
    }
#pragma unroll
    for (int r = 0; r < 8; ++r)
        out[(t0 + r + 8 * half) * N_OUT + n0 + l] = c[r];
}

// ---------------------------------------------------------------------------
extern "C" void kernel_launch(void* const* d_in, const int* in_sizes, int n_in,
                              void* d_out, int out_size, void* d_ws, size_t ws_size,
                              hipStream_t stream) {
    const float* X    = (const float*)d_in[0];   // [T, 32]
    const float* Y    = (const float*)d_in[1];   // [T, 32]
    const float* Win  = (const float*)d_in[2];   // [4000, 32]
    const float* Wres = (const float*)d_in[3];   // [4000, 4000]
    const float* Wfb  = (const float*)d_in[4];   // [4000, 32]
    const float* Wout = (const float*)d_in[5];   // [32, 4032]
    const float* b    = (const float*)d_in[6];   // scalar
    float* out = (float*)d_out;                  // [T, 32]

    // workspace layout
    float* drive  = (float*)d_ws;                              // T*N f32
    float* states = drive  + (size_t)T_STEPS * N_NODES;        // T*N f32
    float* h0     = states + (size_t)T_STEPS * N_NODES;        // 4096 f32
    float* h1     = h0 + 4096;                                 // 4096 f32
    unsigned* bar = (unsigned*)(h1 + 4096);                    // {count, gen}
    size_t base_end = (size_t)((char*)(bar + 2) - (char*)d_ws);
    base_end = (base_end + 15) & ~(size_t)15;
    unsigned* Wbf = (unsigned*)((char*)d_ws + base_end);       // 8M dwords
    const size_t bf16_end = base_end + (size_t)(N_NODES / 2) * N_NODES * 4;

    if (ws_size < base_end) return;                            // ws too small
    const bool use_bf16 = (ws_size >= bf16_end);

    // zero h ping-pong buffers + barrier words (capture-legal memset node)
    hipMemsetAsync(h0, 0, 2 * 4096 * sizeof(float) + 2 * sizeof(unsigned), stream);

    // Phase 1: 128*250 tiles, 1 wave each, 8 waves/block -> 4000 blocks
    drive_gemm_kernel<<<dim3((128 * 250) / 8), dim3(256), 0, stream>>>(
        X, Y, Win, Wfb, b, drive);

    // Phase 2: persistent sequential scan (bf16 weights when ws permits)
    if (use_bf16) {
        wres_to_bf16_kernel<<<dim3(15625), dim3(256), 0, stream>>>(Wres, Wbf);
        reservoir_kernel<true><<<dim3(RES_BLOCKS), dim3(RES_THREADS), 0, stream>>>(
            Wres, Wbf, drive, h0, h1, states, bar + 0, bar + 1);
    } else {
        reservoir_kernel<false><<<dim3(RES_BLOCKS), dim3(RES_THREADS), 0, stream>>>(
            Wres, Wbf, drive, h0, h1, states, bar + 0, bar + 1);
    }

    // Phase 3: 128*2 tiles, 1 wave each -> 32 blocks of 256
    out_gemm_kernel<<<dim3((128 * 2) / 8), dim3(256), 0, stream>>>(
        X, states, Wout, out);

    (void)in_sizes; (void)n_in; (void)out_size;
}